// LengthRegulator_8375186227377
// MI455X (gfx1250) — compile-verified
//
#include <hip/hip_runtime.h>

// LengthRegulator for MI455X (gfx1250, wave32).
//
// Roofline: ~101 MB of HBM traffic, ~0 FLOPs -> pure store-bandwidth problem
// (~4.3 us at 23.3 TB/s). WMMA would only add traffic (materializing a one-hot
// alignment matrix) and precision loss, so the optimal path is wave32 b128
// copies with non-temporal store hints + the gfx1250 async/prefetch paths.

typedef __attribute__((ext_vector_type(4))) float f32x4;

#define LR_N 16
#define LR_L 512
#define LR_D 384      // 384 floats = 96 float4 = 3 float4 per lane of a wave32
#define LR_T 3584     // mel_max_length (== 512 * 7, max possible total)
#define FRAMES_PER_BLOCK 8   // 8 waves of 32 per block, one frame per wave

// ---------------------------------------------------------------------------
// Kernel 1: per-row inclusive scan of durations, invert cumsum into a
// frame->token map, mark invalid tail frames with -1, and emit the durations
// pass-through (2nd tuple output) bit-exactly.
//
// gfx1250 specifics:
//  * durations staged global->LDS with GLOBAL_LOAD_ASYNC_TO_LDS_B32 (ASYNCcnt
//    path); each lane copies its own 4 bytes, so s_wait_asynccnt 0 alone makes
//    the wave's own data visible (no barrier needed before the scan).
//  * scan is wave32-native: 5 __shfl_up steps in-register per wave, then a
//    single 16-element cross-wave scan by wave 0 -> only 2 workgroup barriers
//    (vs 18 for an LDS Hillis-Steele over 512).
// Grid: LR_N blocks x LR_L threads (16 waves).
// ---------------------------------------------------------------------------
__global__ __launch_bounds__(LR_L) void lr_build_idx(
    const int* __restrict__ durations,   // [N, L]
    int* __restrict__ idxmap,            // [N, T] (workspace)
    int* __restrict__ dur_out) {         // [N, L] tail of d_out (int32 bits)
  const int n    = blockIdx.x;
  const int j    = threadIdx.x;          // token index 0..L-1
  const int lane = j & 31;
  const int wave = j >> 5;               // 0..15

  __shared__ int sdur[LR_L];
  __shared__ int wsum[LR_L / 32];        // per-wave inclusive totals

  // --- async global -> LDS stage of this lane's duration -------------------
  {
    const unsigned lds_off = (unsigned)(unsigned long long)(&sdur[j]);
    const int* gaddr = durations + n * LR_L + j;
    asm volatile("global_load_async_to_lds_b32 %0, %1, off"
                 :: "v"(lds_off), "v"(gaddr)
                 : "memory");
    asm volatile("s_wait_asynccnt 0x0" ::: "memory");
  }
  const int d = sdur[j];

  // --- wave32 inclusive scan (in-register) ---------------------------------
  int s = d;
  #pragma unroll
  for (int off = 1; off < 32; off <<= 1) {
    const int v = __shfl_up(s, off, 32);
    if (lane >= off) s += v;
  }

  if (lane == 31) wsum[wave] = s;        // wave totals
  __syncthreads();

  // --- scan the 16 wave totals with wave 0 ---------------------------------
  if (wave == 0) {
    int ws = (lane < (LR_L / 32)) ? wsum[lane] : 0;
    #pragma unroll
    for (int off = 1; off < (LR_L / 32); off <<= 1) {
      const int v = __shfl_up(ws, off, 32);
      if (lane >= off) ws += v;
    }
    if (lane < (LR_L / 32)) wsum[lane] = ws;   // inclusive wave-prefix
  }
  __syncthreads();

  const int base  = (wave == 0) ? 0 : wsum[wave - 1];
  const int end   = base + s;            // cum[j]   (inclusive)
  const int start = end - d;             // cum[j-1]
  const int total = wsum[(LR_L / 32) - 1];

  int* __restrict__ row = idxmap + n * LR_T;

  // Scatter: frames [start, end) belong to token j (duration <= 7).
  for (int t = start; t < end; ++t) row[t] = j;

  // Frames past the total duration are invalid -> -1 (disjoint range, no race).
  for (int t = total + j; t < LR_T; t += LR_L) row[t] = -1;

  // Tuple output #2: durations passed through (raw int32 bits into d_out tail).
  dur_out[n * LR_L + j] = d;
}

// ---------------------------------------------------------------------------
// Kernel 2: the bandwidth kernel. One wave32 per output frame; each lane moves
// 3 x float4 so a wave covers the full 1536-byte D-row with coalesced
// global_load_b128 / global_store_b128 (stores carry the NT temporal hint:
// the 88 MB output is never re-read).
// Grid: (N*T)/FRAMES_PER_BLOCK blocks x 256 threads.
// ---------------------------------------------------------------------------
__global__ __launch_bounds__(256) void lr_expand(
    const float* __restrict__ x,         // [N, L, D]
    const int* __restrict__ idxmap,      // [N, T]
    float* __restrict__ out) {           // [N, T, D]
  const int lane  = threadIdx.x & 31;
  const int frame = blockIdx.x * FRAMES_PER_BLOCK + (threadIdx.x >> 5);
  const int n = frame / LR_T;
  const int t = frame - n * LR_T;

  // All 32 lanes read the same map entry -> single coalesced request.
  const int j = idxmap[n * LR_T + t];

  f32x4* __restrict__ dst = (f32x4*)(out + (size_t)frame * LR_D);

  if (j < 0) {
    // Past total duration: write zeros (keeps output fully defined each call).
    const f32x4 z = {0.f, 0.f, 0.f, 0.f};
    #pragma unroll
    for (int k = 0; k < 3; ++k)
      __builtin_nontemporal_store(z, dst + k * 32 + lane);
  } else {
    const f32x4* __restrict__ src =
        (const f32x4*)(x + ((size_t)n * LR_L + j) * LR_D);
    // gfx1250 prefetch path (global_prefetch_b8); src row is L2-resident
    // (whole x = 12.6 MB << 192 MB L2) so this mostly warms L0.
    __builtin_prefetch(src + lane, 0, 0);
    #pragma unroll
    for (int k = 0; k < 3; ++k) {
      const f32x4 v = src[k * 32 + lane];
      __builtin_nontemporal_store(v, dst + k * 32 + lane);
    }
  }
}

// ---------------------------------------------------------------------------
// Host-side launcher. Inputs: d_in[0]=x (f32 [16,512,384]), d_in[1]=durations
// (i32 [16,512]), d_in[2]=mel_max_length scalar (fixed 3584 in the reference).
// d_out = [N*T*D floats | N*L int32 duration bits]. d_ws holds the idx map
// (16*3584*4 = 229 KB).
// ---------------------------------------------------------------------------
extern "C" void kernel_launch(void* const* d_in, const int* in_sizes, int n_in,
                              void* d_out, int out_size, void* d_ws, size_t ws_size,
                              hipStream_t stream) {
  (void)in_sizes; (void)n_in; (void)out_size; (void)ws_size;

  const float* x         = (const float*)d_in[0];
  const int*   durations = (const int*)d_in[1];

  int*   idxmap  = (int*)d_ws;
  float* out     = (float*)d_out;
  int*   dur_out = (int*)d_out + (size_t)LR_N * LR_T * LR_D;

  lr_build_idx<<<LR_N, LR_L, 0, stream>>>(durations, idxmap, dur_out);
  lr_expand<<<(LR_N * LR_T) / FRAMES_PER_BLOCK, 256, 0, stream>>>(x, idxmap, out);
}